// GNNRefinement_80195629351383
// MI455X (gfx1250) — compile-verified
//
#include <hip/hip_runtime.h>

#define N_NODES 50000
#define N_EDGES 800000
#define D 128
#define N_LAYERS 3
#define LN_EPS 1e-5f

#define ROWS 32      // rows per fused block
#define LDSP 132     // padded LDS row stride (floats): 132 % 64 = 4 -> conflict-free col access

typedef __attribute__((ext_vector_type(2))) float v2f;
typedef __attribute__((ext_vector_type(8))) float v8f;

// ---------------------------------------------------------------------------
// Zero the aggregation numerator + degree arrays (contiguous in workspace).
// ---------------------------------------------------------------------------
__global__ void k_zero(float* __restrict__ p, int n) {
    int i = blockIdx.x * blockDim.x + threadIdx.x;
    const int stride = gridDim.x * blockDim.x;
    for (; i < n; i += stride) p[i] = 0.0f;
}

// ---------------------------------------------------------------------------
// Edge scatter: one wave32 per edge. Lane l owns features [4l, 4l+4).
// agg[dst] += h[src]; deg[dst] += 1.
// ---------------------------------------------------------------------------
__global__ __launch_bounds__(256) void k_scatter(const float* __restrict__ h,
                                                 const long long* __restrict__ ei,
                                                 float* __restrict__ agg,
                                                 float* __restrict__ deg) {
    const int lane = threadIdx.x & 31;
    const int wave = threadIdx.x >> 5;
    const int e = blockIdx.x * 8 + wave;     // 8 waves per block
    if (e >= N_EDGES) return;
    const int src = (int)ei[e];
    const int dst = (int)ei[N_EDGES + e];
    const float4 v = *reinterpret_cast<const float4*>(h + (size_t)src * D + lane * 4);
    float* a = agg + (size_t)dst * D + lane * 4;
    atomicAdd(a + 0, v.x);
    atomicAdd(a + 1, v.y);
    atomicAdd(a + 2, v.z);
    atomicAdd(a + 3, v.w);
    if (lane == 0) atomicAdd(deg + dst, 1.0f);
}

// ---------------------------------------------------------------------------
// Fused: mean = agg/max(deg,1); out = mean@Wl^T + bl + h@Wr^T;
// h_out = h + relu(layernorm(out)).
// Block: 512 threads = 16 waves, covers 32 node rows x 128 output cols.
// Wave w: row-tile (w>>3), col-tile (w&7); one 16x16 f32 tile per wave via
// V_WMMA_F32_16X16X4_F32, K=128 per GEMM (32 wmma x 2 GEMMs).
// ---------------------------------------------------------------------------
__global__ __launch_bounds__(512) void k_fused(const float* __restrict__ h_in,
                                               const float* __restrict__ agg,
                                               const float* __restrict__ deg,
                                               const float* __restrict__ Wl,
                                               const float* __restrict__ bl,
                                               const float* __restrict__ Wr,
                                               const float* __restrict__ gamma,
                                               const float* __restrict__ beta,
                                               float* __restrict__ h_out) {
    __shared__ float s_mean[ROWS * LDSP];
    __shared__ float s_h[ROWS * LDSP];
    __shared__ float s_out[ROWS * LDSP];

    const int tid = threadIdx.x;
    const int lane = tid & 31;
    const int wave = tid >> 5;

    // ---- stage h and mean(=agg/deg) rows into LDS ----
    {
        const int r = tid >> 4;              // 0..31
        const int base = (tid & 15) * 8;     // 0,8,...,120
        const int g = blockIdx.x * ROWS + r;
        if (g < N_NODES) {
            const float inv = 1.0f / fmaxf(deg[g], 1.0f);
            const float4* ap = reinterpret_cast<const float4*>(agg + (size_t)g * D + base);
            const float4* hp = reinterpret_cast<const float4*>(h_in + (size_t)g * D + base);
            float4 a0 = ap[0], a1 = ap[1];
            float4 h0 = hp[0], h1 = hp[1];
            a0.x *= inv; a0.y *= inv; a0.z *= inv; a0.w *= inv;
            a1.x *= inv; a1.y *= inv; a1.z *= inv; a1.w *= inv;
            *reinterpret_cast<float4*>(&s_mean[r * LDSP + base])     = a0;
            *reinterpret_cast<float4*>(&s_mean[r * LDSP + base + 4]) = a1;
            *reinterpret_cast<float4*>(&s_h[r * LDSP + base])        = h0;
            *reinterpret_cast<float4*>(&s_h[r * LDSP + base + 4])    = h1;
        } else {
            const float4 z = make_float4(0.f, 0.f, 0.f, 0.f);
            *reinterpret_cast<float4*>(&s_mean[r * LDSP + base])     = z;
            *reinterpret_cast<float4*>(&s_mean[r * LDSP + base + 4]) = z;
            *reinterpret_cast<float4*>(&s_h[r * LDSP + base])        = z;
            *reinterpret_cast<float4*>(&s_h[r * LDSP + base + 4])    = z;
        }
    }
    __syncthreads();

    // ---- WMMA GEMM: wave owns 16x16 tile ----
    {
        const int rt = wave >> 3;            // row tile within block (0..1)
        const int ct = wave & 7;             // col tile (0..7)
        const int mn = lane & 15;            // M for A-frag, N for B-frag
        const int kb = (lane >> 4) * 2;      // f32 16x4 A layout: lanes>=16 hold K=2,3

        const float* Amean = s_mean + (rt * 16 + mn) * LDSP;
        const float* Ah    = s_h    + (rt * 16 + mn) * LDSP;
        // B[k][n] = W[n][k]: lane n reads contiguous float2 from W row n.
        const float* Brl = Wl + (size_t)(ct * 16 + mn) * D;
        const float* Brr = Wr + (size_t)(ct * 16 + mn) * D;

        v8f acc = {};
#pragma unroll
        for (int k = 0; k < D; k += 4) {
            v2f a = *reinterpret_cast<const v2f*>(Amean + k + kb);
            v2f b = *reinterpret_cast<const v2f*>(Brl + k + kb);
            acc = __builtin_amdgcn_wmma_f32_16x16x4_f32(false, a, false, b,
                                                        (short)0, acc, false, false);
        }
#pragma unroll
        for (int k = 0; k < D; k += 4) {
            v2f a = *reinterpret_cast<const v2f*>(Ah + k + kb);
            v2f b = *reinterpret_cast<const v2f*>(Brr + k + kb);
            acc = __builtin_amdgcn_wmma_f32_16x16x4_f32(false, a, false, b,
                                                        (short)0, acc, false, false);
        }

        // C/D layout: VGPR r -> row (r + 8*(lane>=16)), col = lane&15
        const int orow = rt * 16 + 8 * (lane >> 4);
        const int ocol = ct * 16 + mn;
#pragma unroll
        for (int r = 0; r < 8; ++r)
            s_out[(orow + r) * LDSP + ocol] = acc[r];
    }
    __syncthreads();

    // ---- epilogue: +bl, LayerNorm, ReLU, residual. Wave handles 2 rows. ----
    {
        const float4 blv = *reinterpret_cast<const float4*>(bl + lane * 4);
        const float4 gv  = *reinterpret_cast<const float4*>(gamma + lane * 4);
        const float4 bv  = *reinterpret_cast<const float4*>(beta + lane * 4);
#pragma unroll
        for (int rr = 0; rr < 2; ++rr) {
            const int r = wave * 2 + rr;
            const int g = blockIdx.x * ROWS + r;
            float4 x = *reinterpret_cast<float4*>(&s_out[r * LDSP + lane * 4]);
            x.x += blv.x; x.y += blv.y; x.z += blv.z; x.w += blv.w;

            float sum = x.x + x.y + x.z + x.w;
#pragma unroll
            for (int m = 16; m >= 1; m >>= 1) sum += __shfl_xor(sum, m, 32);
            const float mu = sum * (1.0f / (float)D);

            const float d0 = x.x - mu, d1 = x.y - mu, d2 = x.z - mu, d3 = x.w - mu;
            float s2 = d0 * d0 + d1 * d1 + d2 * d2 + d3 * d3;
#pragma unroll
            for (int m = 16; m >= 1; m >>= 1) s2 += __shfl_xor(s2, m, 32);
            const float rs = rsqrtf(s2 * (1.0f / (float)D) + LN_EPS);

            const float4 hres = *reinterpret_cast<float4*>(&s_h[r * LDSP + lane * 4]);
            float4 o;
            o.x = hres.x + fmaxf(d0 * rs * gv.x + bv.x, 0.0f);
            o.y = hres.y + fmaxf(d1 * rs * gv.y + bv.y, 0.0f);
            o.z = hres.z + fmaxf(d2 * rs * gv.z + bv.z, 0.0f);
            o.w = hres.w + fmaxf(d3 * rs * gv.w + bv.w, 0.0f);
            if (g < N_NODES)
                *reinterpret_cast<float4*>(h_out + (size_t)g * D + lane * 4) = o;
        }
    }
}

// ---------------------------------------------------------------------------
// Host launcher.
// Inputs: 0=node_features f32[50000,128], 1=edge_index i64[2,800000],
//         2=Wl f32[3,128,128], 3=bl f32[3,128], 4=Wr f32[3,128,128],
//         5=gamma f32[3,128], 6=beta f32[3,128]. Out: f32[50000,128].
// Workspace: h_work[50000*128] | agg[50000*128] | deg[50000]  (~51.4 MB)
// ---------------------------------------------------------------------------
extern "C" void kernel_launch(void* const* d_in, const int* in_sizes, int n_in,
                              void* d_out, int out_size, void* d_ws, size_t ws_size,
                              hipStream_t stream) {
    const float* nf        = (const float*)d_in[0];
    const long long* ei    = (const long long*)d_in[1];
    const float* Wl        = (const float*)d_in[2];
    const float* bl        = (const float*)d_in[3];
    const float* Wr        = (const float*)d_in[4];
    const float* gamma     = (const float*)d_in[5];
    const float* beta      = (const float*)d_in[6];
    float* out             = (float*)d_out;

    float* h_work = (float*)d_ws;
    float* agg    = h_work + (size_t)N_NODES * D;
    float* deg    = agg + (size_t)N_NODES * D;

    const int nzero = N_NODES * D + N_NODES;   // agg and deg are contiguous
    const int zero_blocks = (nzero + 255) / 256;
    const int scat_blocks = (N_EDGES + 7) / 8;
    const int fuse_blocks = (N_NODES + ROWS - 1) / ROWS;

    for (int layer = 0; layer < N_LAYERS; ++layer) {
        const float* h_in = (layer == 0) ? nf : h_work;
        float* h_o = (layer == N_LAYERS - 1) ? out : h_work;  // in-place OK: blocks only write their own rows

        k_zero<<<zero_blocks, 256, 0, stream>>>(agg, nzero);
        k_scatter<<<scat_blocks, 256, 0, stream>>>(h_in, ei, agg, deg);
        k_fused<<<fuse_blocks, 512, 0, stream>>>(h_in, agg, deg,
                                                 Wl + (size_t)layer * D * D,
                                                 bl + (size_t)layer * D,
                                                 Wr + (size_t)layer * D * D,
                                                 gamma + (size_t)layer * D,
                                                 beta + (size_t)layer * D,
                                                 h_o);
    }
}